// Encoder_31533649887780
// MI455X (gfx1250) — compile-verified
//
#include <hip/hip_runtime.h>
#include <hip/hip_bf16.h>

// ---------------------------------------------------------------------------
// Constants (match reference)
// ---------------------------------------------------------------------------
#define NN      50000
#define EE      600000
#define ETOT    (EE + NN)     // edges + self loops
#define F_IN    5
#define HH      128
#define GG      64
#define NEG_SLOPE 0.2f
#define BN_EPS  1e-5f

typedef float v2f __attribute__((ext_vector_type(2)));
typedef float v8f __attribute__((ext_vector_type(8)));
typedef unsigned int u32x4 __attribute__((ext_vector_type(4)));
typedef int i32x4 __attribute__((ext_vector_type(4)));
typedef int i32x8 __attribute__((ext_vector_type(8)));

// ---------------------------------------------------------------------------
// Helpers: monotonic float <-> uint mapping for atomic segment-max
// ---------------------------------------------------------------------------
__device__ __forceinline__ unsigned floatFlip(float f) {
    unsigned u = __float_as_uint(f);
    return u ^ ((unsigned)(-(int)(u >> 31)) | 0x80000000u);
}
__device__ __forceinline__ float floatUnflip(unsigned u) {
    u ^= (((u >> 31) - 1u) | 0x80000000u);
    return __uint_as_float(u);
}

__device__ __forceinline__ void edge_pair(const long long* __restrict__ ei,
                                          int e, int& s, int& d) {
    if (e < EE) { s = (int)ei[e]; d = (int)ei[EE + e]; }
    else        { s = e - EE; d = e - EE; }       // self loop
}

// ---------------------------------------------------------------------------
// Generic zero-fill (float-sized words)
// ---------------------------------------------------------------------------
__global__ void k_zero(float* __restrict__ p, int n) {
    int i = blockIdx.x * blockDim.x + threadIdx.x;
    if (i < n) p[i] = 0.0f;
}

// ---------------------------------------------------------------------------
// Layer 1 GEMM: h[N,128] = x[N,5] @ W1[5,128]   (K too small for WMMA)
// ---------------------------------------------------------------------------
__global__ void k_gemm_l1(const float* __restrict__ x, const float* __restrict__ W,
                          float* __restrict__ h) {
    int i = blockIdx.x * blockDim.x + threadIdx.x;   // over N*H
    if (i >= NN * HH) return;
    int n = i >> 7, c = i & 127;
    float s = 0.f;
#pragma unroll
    for (int k = 0; k < F_IN; ++k) s += x[n * F_IN + k] * W[k * HH + c];
    h[i] = s;
}

// ---------------------------------------------------------------------------
// Layers 2/3 GEMM via V_WMMA_F32_16X16X4_F32:
//   C[N,128] = A[N,128] @ W[128,128]
// Block = 256 threads = 8 waves; block -> 80-row tile (50000 = 625*80 exact),
// wave w -> cols [16w,16w+16).  Each wave keeps 5 accumulators (5 row tiles),
// so each B fragment load feeds 5 WMMAs (5x B reuse vs 1 tile/wave).
// A tile (80x128 f32) staged in LDS with padded row stride 132 floats,
// either via the Tensor Data Mover (D# pad fields reproduce the padding)
// or a cooperative VALU fallback.  EXEC stays all-ones for all WMMAs.
// ---------------------------------------------------------------------------
#define ASTRIDE 132           // 128 + 4 DWORD pad per row (bank-conflict free)
#define RT      5             // row tiles per block (5*16 = 80 rows)

__global__ void k_gemm_wmma(const float* __restrict__ A, const float* __restrict__ W,
                            float* __restrict__ C) {
    __shared__ float As[RT * 16 * ASTRIDE];        // 42240 B; ONLY lds object
    const int row0 = blockIdx.x * (RT * 16);
    const int wave = threadIdx.x >> 5;
    const int lane = threadIdx.x & 31;
    const int hlf  = lane >> 4;        // lane half: K offset += 2*hlf
    const int l    = lane & 15;        // row (A) / col (B,C) within tile
    const int col0 = wave * 16;

#if __has_builtin(__builtin_amdgcn_tensor_load_to_lds)
    // ---- TDM: one DMA of the 80x128 tile, global -> LDS, padded stride ----
    if (threadIdx.x < 32) {
        unsigned long long gaddr =
            (unsigned long long)(const void*)A + (unsigned long long)row0 * (HH * 4);
        u32x4 g0;
        g0[0] = 1u;                              // count=1, user descriptor
        g0[1] = 0u;                              // lds_addr = 0 (As is at base)
        g0[2] = (unsigned)(gaddr & 0xFFFFFFFFull);
        g0[3] = (unsigned)(gaddr >> 32) | (2u << 30);   // type=2 ("image")
        i32x8 g1;
        g1[0] = (2 << 16)      // data_size = 4 bytes
              | (1 << 20)      // pad_enable
              | (6 << 22)      // pad_interval: every 128 DWORDs (one row)
              | (3 << 25);     // pad_amount: 4 DWORDs -> 132-float row stride
        g1[1] = (HH & 0xFFFF) << 16;             // tensor_dim0 = 128 (lo16)
        g1[2] = (RT * 16) << 16;                 // dim0 hi16=0 | tensor_dim1=80 lo16
        g1[3] = (HH & 0xFFFF) << 16;             // dim1 hi16=0 | tile_dim0 = 128
        g1[4] = (RT * 16);                       // tile_dim1 = 80, tile_dim2 = 0
        g1[5] = HH;                              // tensor_dim0_stride = 128 (lo32)
        g1[6] = 0;                               // stride hi16 | dim1_stride lo16
        g1[7] = 0;
        i32x4 gz4 = {0, 0, 0, 0};                // groups 2/3 unused (2D tile)
        i32x8 gz8 = {0, 0, 0, 0, 0, 0, 0, 0};
        // amdgpu-toolchain (clang-23) 6-arg form:
        // (u32x4 g0, i32x8 g1, i32x4, i32x4, i32x8, i32 cpol)
        __builtin_amdgcn_tensor_load_to_lds(g0, g1, gz4, gz4, gz8, 0);
    }
    __builtin_amdgcn_s_wait_tensorcnt(0);        // own-wave counter (0 if non-issuer)
    // The TDM writes As through the D# descriptor, which LLVM cannot see.
    // Escape As into an asm with memory effects so the compiler must treat
    // it as written (otherwise the As loads fold to undef and vanish).
    asm volatile("" : : "r"(&As[0]) : "memory");
#else
    // ---- fallback: cooperative, coalesced load of the A tile ----
    for (int i = threadIdx.x; i < RT * 16 * HH; i += 256) {
        int r = i >> 7, c = i & 127;
        As[r * ASTRIDE + c] = A[(size_t)(row0 + r) * HH + c];
    }
#endif
    __syncthreads();

    v8f acc[RT];
#pragma unroll
    for (int t = 0; t < RT; ++t) acc[t] = (v8f){};

#pragma unroll 2
    for (int k = 0; k < HH; k += 4) {
        // B 4x16 fragment (rows K{0,2}/K{1,3} striped across lane halves)
        v2f b;
        b.x = W[(size_t)(k + 2 * hlf + 0) * HH + col0 + l];
        b.y = W[(size_t)(k + 2 * hlf + 1) * HH + col0 + l];
#pragma unroll
        for (int t = 0; t < RT; ++t) {
            // A 16x4 fragment: VGPR0 = K{0,2}, VGPR1 = K{1,3} per lane half
            const float* ar = &As[(t * 16 + l) * ASTRIDE + k + 2 * hlf];
            v2f a; a.x = ar[0]; a.y = ar[1];
            acc[t] = __builtin_amdgcn_wmma_f32_16x16x4_f32(
                /*neg_a=*/false, a, /*neg_b=*/false, b,
                /*c_mod=*/(short)0, acc[t], /*reuse_a=*/false, /*reuse_b=*/false);
        }
    }

    // C/D layout: VGPR i holds rows i (lanes 0-15) and i+8 (lanes 16-31)
#pragma unroll
    for (int t = 0; t < RT; ++t) {
        float* Cout = C + (size_t)(row0 + t * 16) * HH + col0;
#pragma unroll
        for (int i = 0; i < 8; ++i)
            Cout[(size_t)(i + 8 * hlf) * HH + l] = acc[t][i];
    }
}

// ---------------------------------------------------------------------------
// Per-node attention scalars: as[n] = h[n]·a_src, ad[n] = h[n]·a_dst
// ---------------------------------------------------------------------------
__global__ void k_alphas(const float* __restrict__ h,
                         const float* __restrict__ a_src, const float* __restrict__ a_dst,
                         float* __restrict__ as_, float* __restrict__ ad_) {
    int n = blockIdx.x * blockDim.x + threadIdx.x;
    if (n >= NN) return;
    const float* hp = h + (size_t)n * HH;
    float s = 0.f, d = 0.f;
#pragma unroll 8
    for (int k = 0; k < HH; ++k) { float v = hp[k]; s += v * a_src[k]; d += v * a_dst[k]; }
    as_[n] = s; ad_[n] = d;
}

// ---------------------------------------------------------------------------
// Edge pass 1: logits + segment max (flipped-uint atomicMax)
// ---------------------------------------------------------------------------
__global__ void k_edge_logits(const long long* __restrict__ ei,
                              const float* __restrict__ as_, const float* __restrict__ ad_,
                              float* __restrict__ logit, unsigned* __restrict__ mbits) {
    int e = blockIdx.x * blockDim.x + threadIdx.x;
    if (e >= ETOT) return;
    int s, d; edge_pair(ei, e, s, d);
    float z = as_[s] + ad_[d];
    z = z > 0.f ? z : NEG_SLOPE * z;          // leaky relu
    logit[e] = z;
    atomicMax(&mbits[d], floatFlip(z));
}

// ---------------------------------------------------------------------------
// Edge pass 2: ex = exp(logit - max[dst]); denom[dst] += ex  (logit -> ex)
// ---------------------------------------------------------------------------
__global__ void k_edge_exp(const long long* __restrict__ ei,
                           const unsigned* __restrict__ mbits,
                           float* __restrict__ logit_ex, float* __restrict__ denom) {
    int e = blockIdx.x * blockDim.x + threadIdx.x;
    if (e >= ETOT) return;
    int s, d; edge_pair(ei, e, s, d);
    float m  = floatUnflip(mbits[d]);
    float ex = __expf(logit_ex[e] - m);
    logit_ex[e] = ex;
    atomicAdd(&denom[d], ex);
}

// ---------------------------------------------------------------------------
// Init output to bias (out later accumulates segment sum)
// ---------------------------------------------------------------------------
__global__ void k_init_bias(float* __restrict__ out, const float* __restrict__ b) {
    int i = blockIdx.x * blockDim.x + threadIdx.x;
    if (i < NN * HH) out[i] = b[i & 127];
}

// ---------------------------------------------------------------------------
// Edge pass 3: out[dst] += (ex/denom[dst]) * h[src]; one wave per edge,
// float4 per lane (32 lanes * 4 = 128 features).
// ---------------------------------------------------------------------------
__global__ void k_edge_aggr(const long long* __restrict__ ei,
                            const float* __restrict__ ex, const float* __restrict__ denom,
                            const float* __restrict__ h, float* __restrict__ out) {
    int wid  = (blockIdx.x * blockDim.x + threadIdx.x) >> 5;
    int lane = threadIdx.x & 31;
    if (wid >= ETOT) return;
    int s, d; edge_pair(ei, wid, s, d);
    float w = ex[wid] / denom[d];
    float4 v = ((const float4*)(h + (size_t)s * HH))[lane];
    float* o = out + (size_t)d * HH + lane * 4;
    atomicAdd(o + 0, w * v.x);
    atomicAdd(o + 1, w * v.y);
    atomicAdd(o + 2, w * v.z);
    atomicAdd(o + 3, w * v.w);
}

// ---------------------------------------------------------------------------
// ReLU in place
// ---------------------------------------------------------------------------
__global__ void k_relu(float* __restrict__ h) {
    int i = blockIdx.x * blockDim.x + threadIdx.x;
    if (i < NN * HH) h[i] = fmaxf(h[i], 0.f);
}

// ---------------------------------------------------------------------------
// BatchNorm stats: per-column sum / sumsq via LDS accumulation
// ---------------------------------------------------------------------------
__global__ void k_bn_stats(const float* __restrict__ h,
                           float* __restrict__ sum, float* __restrict__ sq) {
    __shared__ float ls[HH], lq[HH];
    if (threadIdx.x < HH) { ls[threadIdx.x] = 0.f; lq[threadIdx.x] = 0.f; }
    __syncthreads();
    const int total = NN * HH;
    for (int i = blockIdx.x * blockDim.x + threadIdx.x; i < total;
         i += gridDim.x * blockDim.x) {
        float v = h[i];
        atomicAdd(&ls[i & 127], v);
        atomicAdd(&lq[i & 127], v * v);
    }
    __syncthreads();
    if (threadIdx.x < HH) {
        atomicAdd(&sum[threadIdx.x], ls[threadIdx.x]);
        atomicAdd(&sq[threadIdx.x],  lq[threadIdx.x]);
    }
}

__global__ void k_bn_apply(float* __restrict__ h,
                           const float* __restrict__ sum, const float* __restrict__ sq,
                           const float* __restrict__ gamma, const float* __restrict__ beta) {
    int i = blockIdx.x * blockDim.x + threadIdx.x;
    if (i >= NN * HH) return;
    int c = i & 127;
    float mu  = sum[c] * (1.0f / NN);
    float var = sq[c] * (1.0f / NN) - mu * mu;
    h[i] = (h[i] - mu) * rsqrtf(var + BN_EPS) * gamma[c] + beta[c];
}

// ---------------------------------------------------------------------------
// Global mean pool: wave per node, accumulate into pool[g], count per graph
// ---------------------------------------------------------------------------
__global__ void k_pool_acc(const float* __restrict__ h, const long long* __restrict__ batch,
                           float* __restrict__ pool, float* __restrict__ cnt) {
    int wid  = (blockIdx.x * blockDim.x + threadIdx.x) >> 5;
    int lane = threadIdx.x & 31;
    if (wid >= NN) return;
    int g = (int)batch[wid];
    float4 v = ((const float4*)(h + (size_t)wid * HH))[lane];
    float* p = pool + (size_t)g * HH + lane * 4;
    atomicAdd(p + 0, v.x);
    atomicAdd(p + 1, v.y);
    atomicAdd(p + 2, v.z);
    atomicAdd(p + 3, v.w);
    if (lane == 0) atomicAdd(&cnt[g], 1.0f);
}

__global__ void k_pool_div(const float* __restrict__ pool, const float* __restrict__ cnt,
                           float* __restrict__ out) {
    int i = blockIdx.x * blockDim.x + threadIdx.x;
    if (i >= GG * HH) return;
    out[i] = pool[i] / fmaxf(cnt[i >> 7], 1.0f);
}

// ---------------------------------------------------------------------------
// Host orchestration
// ---------------------------------------------------------------------------
static inline int cdiv(int a, int b) { return (a + b - 1) / b; }

extern "C" void kernel_launch(void* const* d_in, const int* in_sizes, int n_in,
                              void* d_out, int out_size, void* d_ws, size_t ws_size,
                              hipStream_t stream) {
    // Inputs in setup_inputs() dict order
    const float*     x   = (const float*)d_in[0];
    const long long* ei  = (const long long*)d_in[1];
    const long long* bat = (const long long*)d_in[2];
    const float* W1 = (const float*)d_in[3];
    const float* a1s = (const float*)d_in[4];
    const float* a1d = (const float*)d_in[5];
    const float* b1  = (const float*)d_in[6];
    const float* W2 = (const float*)d_in[7];
    const float* a2s = (const float*)d_in[8];
    const float* a2d = (const float*)d_in[9];
    const float* b2  = (const float*)d_in[10];
    const float* W3 = (const float*)d_in[11];
    const float* a3s = (const float*)d_in[12];
    const float* a3d = (const float*)d_in[13];
    const float* b3  = (const float*)d_in[14];
    const float* g1 = (const float*)d_in[15];
    const float* be1 = (const float*)d_in[16];
    const float* g2 = (const float*)d_in[17];
    const float* be2 = (const float*)d_in[18];

    // Workspace carve-up (floats / 4B words)
    float* ws   = (float*)d_ws;
    float* hA   = ws;                    // N*H  (projected features)
    float* hB   = hA + (size_t)NN * HH;  // N*H  (layer outputs, ping-pong)
    float* as_  = hB + (size_t)NN * HH;  // N
    float* ad_  = as_ + NN;              // N
    unsigned* mb = (unsigned*)(ad_ + NN);// N    (flipped-uint segment max)
    float* den  = (float*)(mb + NN);     // N
    float* exv  = den + NN;              // ETOT (logits -> exp)
    float* sum  = exv + ETOT;            // H
    float* sq   = sum + HH;              // H
    float* pool = sq + HH;               // G*H
    float* cnt  = pool + (size_t)GG * HH;// G
    (void)ws_size; (void)n_in; (void)in_sizes; (void)out_size;

    const int B = 256;
    const int gNH   = cdiv(NN * HH, B);
    const int gN    = cdiv(NN, B);
    const int gE    = cdiv(ETOT, B);
    const int gEagg = cdiv(ETOT * 32, B);
    const int gNagg = cdiv(NN * 32, B);
    const int gTile = NN / (RT * 16);    // 625, exact (50000 = 625*80)

    // ---- one GAT layer ------------------------------------------------
    auto gat_layer = [&](const float* feat_in, bool first,
                         const float* W, const float* asv, const float* adv,
                         const float* bias, float* h, float* out) {
        // zero per-layer scratch (mb, den are adjacent: 2*N words)
        k_zero<<<cdiv(2 * NN, B), B, 0, stream>>>((float*)mb, 2 * NN);
        if (first) k_gemm_l1<<<gNH, B, 0, stream>>>(feat_in, W, h);
        else       k_gemm_wmma<<<gTile, B, 0, stream>>>(feat_in, W, h);
        k_alphas<<<gN, B, 0, stream>>>(h, asv, adv, as_, ad_);
        k_edge_logits<<<gE, B, 0, stream>>>(ei, as_, ad_, exv, mb);
        k_edge_exp<<<gE, B, 0, stream>>>(ei, mb, exv, den);
        k_init_bias<<<gNH, B, 0, stream>>>(out, bias);
        k_edge_aggr<<<gEagg, B, 0, stream>>>(ei, exv, den, h, out);
    };
    auto bn = [&](float* h, const float* gamma, const float* beta) {
        k_zero<<<1, B, 0, stream>>>(sum, 2 * HH);   // sum+sq adjacent
        k_bn_stats<<<512, B, 0, stream>>>(h, sum, sq);
        k_bn_apply<<<gNH, B, 0, stream>>>(h, sum, sq, gamma, beta);
    };

    // Layer 1: x -> hA -> hB ; relu ; bn
    gat_layer(x, true, W1, a1s, a1d, b1, hA, hB);
    k_relu<<<gNH, B, 0, stream>>>(hB);
    bn(hB, g1, be1);

    // Layer 2: hB -> hA -> hB ; relu ; bn
    gat_layer(hB, false, W2, a2s, a2d, b2, hA, hB);
    k_relu<<<gNH, B, 0, stream>>>(hB);
    bn(hB, g2, be2);

    // Layer 3: hB -> hA -> hB (no relu/bn)
    gat_layer(hB, false, W3, a3s, a3d, b3, hA, hB);

    // Global mean pool -> d_out [G, H]
    k_zero<<<cdiv(GG * HH + GG, B), B, 0, stream>>>(pool, GG * HH + GG);
    k_pool_acc<<<gNagg, B, 0, stream>>>(hB, bat, pool, cnt);
    k_pool_div<<<cdiv(GG * HH, B), B, 0, stream>>>(pool, cnt, (float*)d_out);
}